// TaylorMap_74337293959308
// MI455X (gfx1250) — compile-verified
//
#include <hip/hip_runtime.h>

typedef float v2f __attribute__((ext_vector_type(2)));
typedef float v8f __attribute__((ext_vector_type(8)));

#define NF 16           // input features
#define NPOLY 969       // comb(19,3)
#define KPAD 972        // padded to multiple of 4 (243 chunks)
#define TILE_ROWS 16
#define WAVES_PER_WG 8
#define BLOCK_THREADS (WAVES_PER_WG * 32)

// Emit one polynomial feature into the 4-wide staging buffer; every 4th
// feature fires one V_WMMA_F32_16X16X4_F32. 'p' is folded to a constant by
// full unrolling, so f4[] stays in registers and the branch disappears.
#define EMIT(val)                                                            \
  do {                                                                       \
    f4[p & 3] = (val);                                                       \
    if ((p & 3) == 3) {                                                      \
      v2f bfrag = *(const v2f*)(bt + ((p >> 2) << 2));                       \
      v2f afrag;                                                             \
      afrag.x = lo ? f4[0] : f4[2];                                          \
      afrag.y = lo ? f4[1] : f4[3];                                          \
      acc = __builtin_amdgcn_wmma_f32_16x16x4_f32(                           \
          false, afrag, false, bfrag, (short)0, acc, false, false);          \
    }                                                                        \
    ++p;                                                                     \
  } while (0)

__global__ __launch_bounds__(BLOCK_THREADS) void taylor_wmma_kernel(
    const float* __restrict__ X, const float* __restrict__ W,
    float* __restrict__ out, int ntiles) {
  // W transposed in LDS: WT[n*KPAD + k] = W[k*16 + n]; pad rows k>=969 zeroed
  __shared__ float WT[NF * KPAD];  // 62208 bytes
  for (int idx = threadIdx.x; idx < NF * KPAD; idx += BLOCK_THREADS) {
    int k = idx >> 4;
    int n = idx & 15;
    WT[n * KPAD + k] = (k < NPOLY) ? W[k * NF + n] : 0.0f;
  }
  __syncthreads();

  const int wave = threadIdx.x >> 5;
  const int lane = threadIdx.x & 31;
  const int tile = blockIdx.x * WAVES_PER_WG + wave;
  if (tile >= ntiles) return;  // wave-uniform; EXEC stays all-ones

  const bool lo = lane < 16;
  const int n = lane & 15;  // column for B/C/D, row-in-tile for A
  const int row = tile * TILE_ROWS + n;

  // This lane's full input row (A-matrix rows are per-lane).
  float xv[NF];
  {
    const float4* xp = (const float4*)(X + row * NF);
    float4 q0 = xp[0], q1 = xp[1], q2 = xp[2], q3 = xp[3];
    xv[0] = q0.x;  xv[1] = q0.y;  xv[2] = q0.z;  xv[3] = q0.w;
    xv[4] = q1.x;  xv[5] = q1.y;  xv[6] = q1.z;  xv[7] = q1.w;
    xv[8] = q2.x;  xv[9] = q2.y;  xv[10] = q2.z; xv[11] = q2.w;
    xv[12] = q3.x; xv[13] = q3.y; xv[14] = q3.z; xv[15] = q3.w;
  }

  // Residual: init C with the X tile in 16x16 f32 C/D layout
  // (VGPR j: lanes 0-15 -> M=j, lanes 16-31 -> M=j+8; N = lane&15).
  const int m0 = lo ? 0 : 8;
  v8f acc;
#pragma unroll
  for (int j = 0; j < 8; ++j)
    acc[j] = X[(tile * TILE_ROWS + m0 + j) * NF + n];

  // B fragment base: lane reads WT[n][4c+koff], WT[n][4c+koff+1] (ds_load_b64)
  const float* bt = &WT[n * KPAD + (lo ? 0 : 2)];

  float f4[4];
  int p = 0;

  // Feature stream in sklearn PolynomialFeatures order (matches reference):
  EMIT(1.0f);  // constant
#pragma unroll
  for (int i = 0; i < NF; ++i) EMIT(xv[i]);  // linear
#pragma unroll
  for (int f = 0; f < NF; ++f) {  // degree 2: x_f * x_j, j >= f
#pragma unroll
    for (int j = f; j < NF; ++j) EMIT(xv[f] * xv[j]);
  }
#pragma unroll
  for (int f = 0; f < NF; ++f) {  // degree 3: x_f * x_j * x_k, f<=j<=k
#pragma unroll
    for (int j = f; j < NF; ++j) {
      float pfj = xv[f] * xv[j];
#pragma unroll
      for (int k = j; k < NF; ++k) EMIT(pfj * xv[k]);
    }
  }
  // pad 969 -> 972 with zeros (WT pad rows are zeroed too)
  EMIT(0.0f);
  EMIT(0.0f);
  EMIT(0.0f);

  // Store D (+X already folded in via C init).
#pragma unroll
  for (int j = 0; j < 8; ++j)
    out[(tile * TILE_ROWS + m0 + j) * NF + n] = acc[j];
}

extern "C" void kernel_launch(void* const* d_in, const int* in_sizes, int n_in,
                              void* d_out, int out_size, void* d_ws,
                              size_t ws_size, hipStream_t stream) {
  const float* X = (const float*)d_in[0];  // [B, 16] f32
  const float* W = (const float*)d_in[1];  // [969, 16] f32
  float* out = (float*)d_out;              // [B, 16] f32

  const int batch = in_sizes[0] / NF;             // 65536
  const int ntiles = batch / TILE_ROWS;           // 4096
  const int blocks = (ntiles + WAVES_PER_WG - 1) / WAVES_PER_WG;  // 512

  taylor_wmma_kernel<<<blocks, BLOCK_THREADS, 0, stream>>>(X, W, out, ntiles);
}